// FinMamba_88021059764656
// MI455X (gfx1250) — compile-verified
//
#include <hip/hip_runtime.h>
#include <math.h>

// ---------------- constants (from reference) ----------------
#define BB    8
#define LL    20
#define NN    1024
#define FF    64
#define HID   64
#define NLVL  2
#define NHEAD 4
#define NSQ   (NN*NN)          // 1048576

typedef __attribute__((ext_vector_type(2))) float v2f;
typedef __attribute__((ext_vector_type(8))) float v8f;

#define WMMA_F32(a, b, c) \
    __builtin_amdgcn_wmma_f32_16x16x4_f32(false, (a), false, (b), (short)0, (c), false, false)

__device__ __forceinline__ float geluf(float x) {
    return 0.5f * x * (1.f + erff(x * 0.70710678118654752f));
}
__device__ __forceinline__ float sigmf(float x) {
    return 1.f / (1.f + expf(-x));
}
__device__ __forceinline__ unsigned ordkey(float x) {
    unsigned u = __float_as_uint(x);
    return (u & 0x80000000u) ? ~u : (u | 0x80000000u);
}

// ---------------- workspace layout (float offsets) ----------------
static const size_t F_M2   = 0;
static const size_t F_XN   = F_M2 + (size_t)BB*LL*FF;
static const size_t F_W    = F_XN + (size_t)BB*NN*FF;
static const size_t F_AM   = F_W  + (size_t)BB*FF*NN;
static const size_t F_EI   = F_AM + (size_t)BB*NN*NN;
static const size_t F_EJ   = F_EI + (size_t)BB*NHEAD*NN;
static const size_t F_FE   = F_EJ + (size_t)BB*NHEAD*NN;
static const size_t F_TH   = F_FE + (size_t)BB*NN*2*HID;
static const size_t F_KEEP = F_TH + BB;
static const size_t F_END  = F_KEEP + BB;

// =================== stage kernels ===================

// m2[b][l][f] = mean over n of s_seq[b][l][n][f]
__global__ void k_mean(const float* __restrict__ s, float* __restrict__ m2) {
    int idx = blockIdx.x * 256 + threadIdx.x;
    if (idx >= BB*LL*FF) return;
    int f = idx & 63;
    int l = (idx >> 6) % LL;
    int b = idx / (LL*FF);
    const float* p = s + (((size_t)b*LL + l)*NN)*FF + f;
    float acc = 0.f;
    for (int n = 0; n < NN; ++n) acc += p[(size_t)n*FF];
    m2[idx] = acc * (1.f/(float)NN);
}

// s_seq_n = transpose(s_seq, (0,2,1,3)) written straight to output
__global__ void k_sseqn(const float* __restrict__ s, float* __restrict__ out) {
    size_t idx = (size_t)blockIdx.x * 256 + threadIdx.x; // B*N*L*F total
    int f = (int)(idx & 63);
    size_t r = idx >> 6;
    int t = (int)(r % LL); r /= LL;
    int n = (int)(r & (NN-1));
    int b = (int)(r >> 10);
    out[idx] = s[(((size_t)b*LL + t)*NN + n)*FF + f];
}

// inception branches + logits + keep -> selk/prefix init
__global__ void k_incep(const float* __restrict__ m2,
                        const float* w1, const float* b1,
                        const float* w2, const float* b2,
                        const float* w3, const float* b3,
                        const float* w4, const float* b4,
                        const float* ow, const float* ob,
                        unsigned* __restrict__ prefix, int* __restrict__ selk,
                        float* __restrict__ keepv) {
    __shared__ float row[LL*FF];
    __shared__ float red[128];
    __shared__ float xcat[32];
    int b = blockIdx.x, tid = threadIdx.x;
    for (int i = tid; i < LL*FF; i += 128) row[i] = m2[b*LL*FF + i];
    __syncthreads();
    for (int c = 0; c < 32; ++c) {
        int br = c >> 3, ch = c & 7;
        float s = 0.f;
        for (int pos = tid; pos < LL*FF; pos += 128) {
            int y = pos >> 6, x = pos & 63;
            float v;
            if (br == 0) {
                v = w1[ch]*row[pos] + b1[ch];
            } else if (br == 1) {
                v = b2[ch];
                for (int dx = 0; dx < 3; ++dx) {
                    int xx = x + dx - 1;
                    if (xx >= 0 && xx < FF) v += w2[ch*3+dx]*row[y*FF+xx];
                }
            } else if (br == 2) {
                v = b3[ch];
                for (int dy = 0; dy < 3; ++dy) {
                    int yy = y + dy - 1;
                    if (yy >= 0 && yy < LL) v += w3[ch*3+dy]*row[yy*FF+x];
                }
            } else {
                v = b4[ch];
                for (int dy = 0; dy < 3; ++dy)
                    for (int dx = 0; dx < 3; ++dx) {
                        int yy = y+dy-1, xx = x+dx-1;
                        if (yy>=0 && yy<LL && xx>=0 && xx<FF)
                            v += w4[ch*9+dy*3+dx]*row[yy*FF+xx];
                    }
            }
            s += geluf(v);
        }
        red[tid] = s; __syncthreads();
        for (int off = 64; off > 0; off >>= 1) {
            if (tid < off) red[tid] += red[tid+off];
            __syncthreads();
        }
        if (tid == 0) xcat[c] = red[0] * (1.f/(float)(LL*FF));
        __syncthreads();
    }
    if (tid == 0) {
        float lg = ob[0];
        for (int c = 0; c < 32; ++c) lg += xcat[c]*ow[c];
        float kp = fminf(fmaxf(0.2f*sigmf(lg), 0.02f), 1.f);
        int kk = (int)ceilf(kp * (float)NSQ);
        kk = max(1, min(kk, NSQ));
        selk[b] = kk; prefix[b] = 0u; keepv[b] = kp;
    }
}

// xn = row-normalized s_seq[:, -1]
__global__ void k_xn(const float* __restrict__ s, float* __restrict__ xn) {
    int bn = blockIdx.x;
    int b = bn >> 10, n = bn & (NN-1);
    int tid = threadIdx.x; // 64
    __shared__ float red[64];
    const float* xr = s + (((size_t)b*LL + (LL-1))*NN + n)*FF;
    float v = xr[tid];
    red[tid] = v*v; __syncthreads();
    for (int off = 32; off > 0; off >>= 1) {
        if (tid < off) red[tid] += red[tid+off];
        __syncthreads();
    }
    float d = fmaxf(sqrtf(red[0]), 1e-12f);
    xn[(size_t)bn*FF + tid] = v / d;
}

// =================== WMMA f32 16x16x4 GEMMs ===================
// lane: lr = lane&15, kh = lane>>4; A/B elem v: K = kbase + 2*kh + v
// C/D elem v: M = m0 + v + 8*kh, N = n0 + lr

// W[b] (64 x 1024) = xn[b]^T (64x1024) @ prior (1024x1024)
// 1x4 register blocking: A fragment reused across 4 N-tiles.
__global__ void k_gemm_w(const float* __restrict__ xn, const float* __restrict__ prior,
                         float* __restrict__ W) {
    int b = blockIdx.y;
    int g = blockIdx.x;               // 4 mtiles * 16 ngroups
    int gn = g & 15, tm = g >> 4;
    int m0 = tm*16, n0 = gn*64;
    int lane = threadIdx.x & 31;
    int lr = lane & 15, kh = lane >> 4;
    const float* A = xn + (size_t)b*NN*FF;   // A[m][k] = xn[k][m]
    v8f acc0 = {}, acc1 = {}, acc2 = {}, acc3 = {};
    for (int kb = 0; kb < NN; kb += 4) {
        int k = kb + 2*kh;
        v2f a;
        a.x = A[(size_t)k*FF + m0 + lr];
        a.y = A[(size_t)(k+1)*FF + m0 + lr];
        const float* P0 = prior + (size_t)k*NN + lr;
        const float* P1 = prior + (size_t)(k+1)*NN + lr;
        v2f b0 = { P0[n0     ], P1[n0     ] };
        v2f b1 = { P0[n0 + 16], P1[n0 + 16] };
        v2f b2 = { P0[n0 + 32], P1[n0 + 32] };
        v2f b3 = { P0[n0 + 48], P1[n0 + 48] };
        acc0 = WMMA_F32(a, b0, acc0);
        acc1 = WMMA_F32(a, b1, acc1);
        acc2 = WMMA_F32(a, b2, acc2);
        acc3 = WMMA_F32(a, b3, acc3);
    }
    float* C = W + (size_t)b*FF*NN;
    #pragma unroll
    for (int v = 0; v < 8; ++v) {
        size_t r = (size_t)(m0 + v + 8*kh)*NN + lr;
        C[r + n0     ] = acc0[v];
        C[r + n0 + 16] = acc1[v];
        C[r + n0 + 32] = acc2[v];
        C[r + n0 + 48] = acc3[v];
    }
}

// amat[b] (1024x1024) = xn[b] (1024x64) @ W[b] (64x1024)
// 2x2 register blocking; A fragments loaded as contiguous b64.
__global__ void k_gemm_amat(const float* __restrict__ xn, const float* __restrict__ W,
                            float* __restrict__ amat) {
    int b = blockIdx.y;
    int g = blockIdx.x;               // 32*32 groups of 2x2 tiles
    int gn = g & 31, gm = g >> 5;
    int m0 = gm*32, n0 = gn*32;
    int lane = threadIdx.x & 31;
    int lr = lane & 15, kh = lane >> 4;
    const float* A  = xn + (size_t)b*NN*FF;
    const float* Bm = W  + (size_t)b*FF*NN;
    v8f c00 = {}, c01 = {}, c10 = {}, c11 = {};
    for (int kb = 0; kb < FF; kb += 4) {
        int k = kb + 2*kh;            // even -> 8B aligned
        v2f a0 = *(const v2f*)(A + (size_t)(m0      + lr)*FF + k);
        v2f a1 = *(const v2f*)(A + (size_t)(m0 + 16 + lr)*FF + k);
        const float* Q0 = Bm + (size_t)k*NN + lr;
        const float* Q1 = Bm + (size_t)(k+1)*NN + lr;
        v2f b0 = { Q0[n0     ], Q1[n0     ] };
        v2f b1 = { Q0[n0 + 16], Q1[n0 + 16] };
        c00 = WMMA_F32(a0, b0, c00);
        c01 = WMMA_F32(a0, b1, c01);
        c10 = WMMA_F32(a1, b0, c10);
        c11 = WMMA_F32(a1, b1, c11);
    }
    float* C = amat + (size_t)b*NN*NN;
    #pragma unroll
    for (int v = 0; v < 8; ++v) {
        size_t r0 = (size_t)(m0      + v + 8*kh)*NN + lr;
        size_t r1 = (size_t)(m0 + 16 + v + 8*kh)*NN + lr;
        C[r0 + n0     ] = c00[v];
        C[r0 + n0 + 16] = c01[v];
        C[r1 + n0     ] = c10[v];
        C[r1 + n0 + 16] = c11[v];
    }
}

// z[b] (1024 x 1280) = alpha[b] @ S[b], gelu, write (B,N,L,F)
// 2x2 register blocking; 32-wide N-group never crosses a time boundary.
__global__ void k_gemm_z(const float* __restrict__ alpha, const float* __restrict__ s,
                         float* __restrict__ z) {
    int b = blockIdx.y;
    int g = blockIdx.x;               // 32 mgroups * 40 ngroups
    int gn = g % 40, gm = g / 40;
    int m0 = gm*32, n0 = gn*32;
    int t  = n0 >> 6;                 // uniform per group (n0%64 in {0,32})
    int f0 = n0 & 63;
    int lane = threadIdx.x & 31;
    int lr = lane & 15, kh = lane >> 4;
    const float* A  = alpha + (size_t)b*NN*NN;
    const float* Sb = s + ((size_t)b*LL + t)*NN*FF;   // S[k][f] = s_seq[b][t][k][f]
    v8f c00 = {}, c01 = {}, c10 = {}, c11 = {};
    for (int kb = 0; kb < NN; kb += 4) {
        int k = kb + 2*kh;            // even -> 8B aligned
        v2f a0 = *(const v2f*)(A + (size_t)(m0      + lr)*NN + k);
        v2f a1 = *(const v2f*)(A + (size_t)(m0 + 16 + lr)*NN + k);
        const float* Q0 = Sb + (size_t)k*FF + lr;
        const float* Q1 = Sb + (size_t)(k+1)*FF + lr;
        v2f b0 = { Q0[f0     ], Q1[f0     ] };
        v2f b1 = { Q0[f0 + 16], Q1[f0 + 16] };
        c00 = WMMA_F32(a0, b0, c00);
        c01 = WMMA_F32(a0, b1, c01);
        c10 = WMMA_F32(a1, b0, c10);
        c11 = WMMA_F32(a1, b1, c11);
    }
    #pragma unroll
    for (int v = 0; v < 8; ++v) {
        int ma = m0 + v + 8*kh;
        int mb = ma + 16;
        size_t ra = (((size_t)b*NN + ma)*LL + t)*FF + lr;
        size_t rb = (((size_t)b*NN + mb)*LL + t)*FF + lr;
        z[ra + f0     ] = geluf(c00[v]);
        z[ra + f0 + 16] = geluf(c01[v]);
        z[rb + f0     ] = geluf(c10[v]);
        z[rb + f0 + 16] = geluf(c11[v]);
    }
}

// =================== exact top-k radix select ===================
__global__ void k_hist_zero(unsigned* __restrict__ hist) {
    hist[blockIdx.x*256 + threadIdx.x] = 0u;
}

__global__ void k_hist_count(const float* __restrict__ amat,
                             const unsigned* __restrict__ prefix,
                             unsigned* __restrict__ hist, int pass) {
    __shared__ unsigned lh[256];
    int tid = threadIdx.x;
    lh[tid] = 0u; __syncthreads();
    size_t idx = (size_t)blockIdx.x * 256 + tid;
    int b = (int)(idx >> 20);      // NN*NN = 2^20 per batch
    unsigned u = ordkey(amat[idx]);
    int shift = 24 - 8*pass;
    bool ok = (pass == 0) ? true : ((u >> (unsigned)(shift + 8)) == prefix[b]);
    if (ok) atomicAdd(&lh[(u >> (unsigned)shift) & 255u], 1u);
    __syncthreads();
    if (lh[tid]) atomicAdd(&hist[b*256 + tid], lh[tid]);
}

__global__ void k_hist_scan(unsigned* __restrict__ hist, unsigned* __restrict__ prefix,
                            int* __restrict__ selk, float* __restrict__ thresh, int pass) {
    if (threadIdx.x != 0) return;
    int b = blockIdx.x;
    unsigned* h = hist + b*256;
    int kk = selk[b];
    unsigned cum = 0; int bin = 0;
    for (int i = 255; i >= 0; --i) {
        unsigned c = h[i];
        if (cum + c >= (unsigned)kk) { bin = i; break; }
        cum += c;
    }
    prefix[b] = (prefix[b] << 8) | (unsigned)bin;
    selk[b] = kk - (int)cum;
    if (pass == 3) {
        unsigned key = prefix[b];
        unsigned bits = (key & 0x80000000u) ? (key & 0x7FFFFFFFu) : ~key;
        thresh[b] = __uint_as_float(bits);
    }
}

// =================== GAT ===================
__global__ void k_wh(const float* __restrict__ s, const float* __restrict__ gat_w,
                     const float* __restrict__ gat_a,
                     float* __restrict__ ei, float* __restrict__ ej) {
    int bn = blockIdx.x;
    int b = bn >> 10, n = bn & (NN-1);
    int tid = threadIdx.x; // 64
    __shared__ float xr[FF], wh[FF];
    xr[tid] = s[(((size_t)b*LL + (LL-1))*NN + n)*FF + tid];
    __syncthreads();
    float acc = 0.f;
    for (int k = 0; k < FF; ++k) acc += xr[k]*gat_w[k*FF + tid];
    wh[tid] = acc;
    __syncthreads();
    if (tid < 2*NHEAD) {
        int h = tid & 3;
        const float* ar = gat_a + h*(2*FF) + ((tid < NHEAD) ? 0 : FF);
        float e = 0.f;
        for (int f = 0; f < FF; ++f) e += wh[f]*ar[f];
        float* dst = (tid < NHEAD) ? ei : ej;
        dst[(b*NHEAD + h)*NN + n] = e;
    }
}

// masked 4-head softmax fused per row; overwrites amat row with alpha row
__global__ void k_alpha(float* __restrict__ amat, const float* __restrict__ ei,
                        const float* __restrict__ ej, const float* __restrict__ thresh) {
    int bi = blockIdx.x;
    int b = bi >> 10, i = bi & (NN-1);
    int tid = threadIdx.x; // 128
    __shared__ float red[128];
    float* rowp = amat + ((size_t)b*NN + i)*NN;
    float th = thresh[b];
    float av[8], acc[8]; bool mk[8];
    #pragma unroll
    for (int r = 0; r < 8; ++r) {
        int j = tid + 128*r;
        av[r] = rowp[j];
        mk[r] = (av[r] >= th) || (j == i);
        acc[r] = 0.f;
    }
    for (int h = 0; h < NHEAD; ++h) {
        float eih = ei[(b*NHEAD + h)*NN + i];
        float ev[8];
        float lmax = -3.402823466e38f;
        #pragma unroll
        for (int r = 0; r < 8; ++r) {
            int j = tid + 128*r;
            float e = eih + ej[(b*NHEAD + h)*NN + j];
            e = (e >= 0.f) ? e : 0.2f*e;
            ev[r] = e;
            if (mk[r]) lmax = fmaxf(lmax, e);
        }
        red[tid] = lmax; __syncthreads();
        for (int off = 64; off > 0; off >>= 1) {
            if (tid < off) red[tid] = fmaxf(red[tid], red[tid+off]);
            __syncthreads();
        }
        float m = red[0];
        __syncthreads();
        float lsum = 0.f;
        #pragma unroll
        for (int r = 0; r < 8; ++r) {
            ev[r] = mk[r] ? expf(ev[r] - m) : 0.f;
            lsum += ev[r];
        }
        red[tid] = lsum; __syncthreads();
        for (int off = 64; off > 0; off >>= 1) {
            if (tid < off) red[tid] += red[tid+off];
            __syncthreads();
        }
        float inv = 0.25f / red[0];
        __syncthreads();
        #pragma unroll
        for (int r = 0; r < 8; ++r) acc[r] += ev[r]*inv;
    }
    #pragma unroll
    for (int r = 0; r < 8; ++r) rowp[tid + 128*r] = acc[r];
}

// =================== level (SSM) kernel, per (b,n,level) ===================
__global__ void k_level(const float* __restrict__ zbuf, const float* __restrict__ snbuf,
                        const float* __restrict__ lvl_w, const float* __restrict__ lvl_b,
                        const float* __restrict__ ssm_w, const float* __restrict__ ssm_b,
                        const float* __restrict__ so_w, const float* __restrict__ so_b,
                        const float* __restrict__ rel_w, const float* __restrict__ rel_b,
                        float* __restrict__ feats) {
    int bn  = blockIdx.x;
    int lvl = blockIdx.y;
    int tid = threadIdx.x; // 192
    __shared__ float P [LL*128];
    __shared__ float X [LL*128];
    __shared__ float PP[LL*192];
    __shared__ float xm[LL];
    __shared__ float g [HID];
    __shared__ float l2[HID];
    const float* sr = snbuf + (size_t)bn*LL*FF;
    const float* zr = zbuf  + (size_t)bn*LL*FF;
    for (int idx = tid; idx < LL*128; idx += 192) {
        int t = idx >> 7, k = idx & 127;
        P[idx] = (k < FF) ? sr[t*FF + k] : zr[t*FF + (k - FF)];
    }
    __syncthreads();
    const float* lw = lvl_w + (size_t)lvl*128*128;
    const float* lb = lvl_b + lvl*128;
    for (int idx = tid; idx < LL*128; idx += 192) {
        int t = idx >> 7, j = idx & 127;
        float s = lb[j];
        for (int k = 0; k < 128; ++k) s += P[t*128 + k]*lw[k*128 + j];
        X[idx] = s;
    }
    __syncthreads();
    if (tid < LL) {
        float s = 0.f;
        for (int k = 0; k < 128; ++k) s += X[tid*128 + k];
        xm[tid] = s * (1.f/128.f);
    }
    __syncthreads();
    const float* sw = ssm_w + (size_t)lvl*128*192;
    const float* sb = ssm_b + lvl*192;
    {
        int j = tid; // < 192
        for (int t = 0; t < LL; ++t) {
            float s = sb[j];
            for (int k = 0; k < 128; ++k) s += X[t*128 + k]*sw[k*192 + j];
            PP[t*192 + j] = s;
        }
    }
    __syncthreads();
    if (tid < HID) {
        float h = 0.f;
        for (int t = 0; t < LL; ++t) {
            float at = sigmf(PP[t*192 + tid]);
            float dr = PP[t*192 + HID + tid]*xm[t];
            h = at*h + dr;
        }
        float o = PP[(LL-1)*192 + 2*HID + tid]*h;
        g[tid] = geluf(o);
    }
    __syncthreads();
    if (tid < HID) {
        const float* owp = so_w + (size_t)lvl*HID*HID;
        float s = so_b[lvl*HID + tid];
        for (int k = 0; k < HID; ++k) s += g[k]*owp[k*HID + tid];
        l2[tid] = s + xm[LL-1];
    }
    __syncthreads();
    if (tid < HID) {
        const float* rw = rel_w + (size_t)lvl*HID*HID;
        float s = rel_b[lvl*HID + tid];
        for (int k = 0; k < HID; ++k) s += l2[k]*rw[k*HID + tid];
        feats[(size_t)bn*(NLVL*HID) + lvl*HID + tid] = s;
    }
}

// y = feats @ fin_w + fin_b
__global__ void k_final(const float* __restrict__ feats, const float* __restrict__ fw,
                        const float* __restrict__ fb, float* __restrict__ y) {
    int i = blockIdx.x*256 + threadIdx.x;
    if (i >= BB*NN) return;
    float s = fb[0];
    for (int k = 0; k < NLVL*HID; ++k) s += feats[(size_t)i*(NLVL*HID) + k]*fw[k];
    y[i] = s;
}

// =================== launcher ===================
extern "C" void kernel_launch(void* const* d_in, const int* in_sizes, int n_in,
                              void* d_out, int out_size, void* d_ws, size_t ws_size,
                              hipStream_t stream) {
    const float* s_seq   = (const float*)d_in[0];
    const float* prior_d = (const float*)d_in[1];
    const float* w1 = (const float*)d_in[2];  const float* b1 = (const float*)d_in[3];
    const float* w2 = (const float*)d_in[4];  const float* b2 = (const float*)d_in[5];
    const float* w3 = (const float*)d_in[6];  const float* b3 = (const float*)d_in[7];
    const float* w4 = (const float*)d_in[8];  const float* b4 = (const float*)d_in[9];
    const float* ow = (const float*)d_in[10]; const float* ob = (const float*)d_in[11];
    const float* gat_w = (const float*)d_in[12];
    const float* gat_a = (const float*)d_in[13];
    const float* lvl_w = (const float*)d_in[14]; const float* lvl_b = (const float*)d_in[15];
    const float* ssm_w = (const float*)d_in[16]; const float* ssm_b = (const float*)d_in[17];
    const float* so_w  = (const float*)d_in[18]; const float* so_b  = (const float*)d_in[19];
    const float* rel_w = (const float*)d_in[20]; const float* rel_b = (const float*)d_in[21];
    const float* fin_w = (const float*)d_in[22]; const float* fin_b = (const float*)d_in[23];

    float* out_y  = (float*)d_out;                       // B*N
    float* out_z  = out_y + (size_t)BB*NN;               // B*N*L*F
    float* out_sn = out_z + (size_t)BB*NN*LL*FF;         // B*N*L*F

    float* ws = (float*)d_ws;
    float* m2     = ws + F_M2;
    float* xn     = ws + F_XN;
    float* W      = ws + F_W;
    float* amat   = ws + F_AM;     // later reused as alpha
    float* ei     = ws + F_EI;
    float* ej     = ws + F_EJ;
    float* feats  = ws + F_FE;
    float* thresh = ws + F_TH;
    float* keepv  = ws + F_KEEP;
    unsigned* hist   = (unsigned*)(ws + F_END);
    unsigned* prefix = hist + BB*256;
    int*      selk   = (int*)(prefix + BB);

    // stage 0: means, transpose-out, inception gate, row-normalize
    k_mean <<<(BB*LL*FF + 255)/256, 256, 0, stream>>>(s_seq, m2);
    k_sseqn<<<(int)(((size_t)BB*NN*LL*FF)/256), 256, 0, stream>>>(s_seq, out_sn);
    k_incep<<<BB, 128, 0, stream>>>(m2, w1,b1, w2,b2, w3,b3, w4,b4, ow, ob,
                                    prefix, selk, keepv);
    k_xn   <<<BB*NN, 64, 0, stream>>>(s_seq, xn);

    // stage 1: a_mat = xn @ (xn^T @ prior_d)  [WMMA f32, register-blocked]
    k_gemm_w   <<<dim3(64, BB),   32, 0, stream>>>(xn, prior_d, W);
    k_gemm_amat<<<dim3(1024, BB), 32, 0, stream>>>(xn, W, amat);

    // stage 2: exact k-th-largest per batch (4-pass radix select)
    for (int pass = 0; pass < 4; ++pass) {
        k_hist_zero <<<BB, 256, 0, stream>>>(hist);
        k_hist_count<<<(int)(((size_t)BB*NSQ)/256), 256, 0, stream>>>(amat, prefix, hist, pass);
        k_hist_scan <<<BB, 32, 0, stream>>>(hist, prefix, selk, thresh, pass);
    }

    // stage 3: GAT attention (alpha overwrites amat rows in place)
    k_wh   <<<BB*NN, 64, 0, stream>>>(s_seq, gat_w, gat_a, ei, ej);
    k_alpha<<<BB*NN, 128, 0, stream>>>(amat, ei, ej, thresh);

    // stage 4: z_seq = gelu(alpha @ S)  [WMMA f32, register-blocked]
    k_gemm_z<<<dim3(32*40, BB), 32, 0, stream>>>(amat, s_seq, out_z);

    // stage 5: per-(b,n,level) SSM pipeline + final projection
    k_level<<<dim3(BB*NN, NLVL), 192, 0, stream>>>(out_z, out_sn,
                                                   lvl_w, lvl_b, ssm_w, ssm_b,
                                                   so_w, so_b, rel_w, rel_b, feats);
    k_final<<<(BB*NN + 255)/256, 256, 0, stream>>>(feats, fin_w, fin_b, out_y);
}